// CTCPhonemeHead_15977278341728
// MI455X (gfx1250) — compile-verified
//
#include <hip/hip_runtime.h>
#include <hip/hip_bf16.h>
#include <math.h>

// ---------------------------------------------------------------------------
// CTC phoneme head for MI455X (gfx1250, wave32).
//   Kernel 1: fused GEMM (64000x512 @ 512x40) + bias + log_softmax via
//             V_WMMA_F32_16X16X4_F32 (fp32 WMMA: GEMM is HBM-bound at
//             ~141MB/23.3TB/s ~ 6us, so fp32 precision is free).
//   Kernel 2: CTC alpha recursion, one workgroup per batch element,
//             one thread per extended state, LDS ping-pong, 1 barrier/step,
//             emission row fetched one step ahead via GLOBAL_LOAD_ASYNC_TO_LDS
//             (ASYNCcnt) when the toolchain exposes it.
//   Kernel 3: mean reduction of per-batch losses.
// ---------------------------------------------------------------------------

typedef __attribute__((ext_vector_type(2))) float v2f;
typedef __attribute__((ext_vector_type(8))) float v8f;

#define NUM_B  32
#define T_LEN  2000
#define H_DIM  512
#define C_DIM  40
#define L_TGT  200
#define S_EXT  (2 * L_TGT + 1)   // 401
#define BLANKC 39
#define NEGF   (-1e30f)

#if defined(__has_builtin)
#if __has_builtin(__builtin_amdgcn_global_load_async_to_lds_b32)
#define HAVE_ASYNC_LDS 1
#endif
#endif
#ifndef HAVE_ASYNC_LDS
#define HAVE_ASYNC_LDS 0
#endif

#if HAVE_ASYNC_LDS
typedef __attribute__((address_space(1))) int g1_int;   // clang prints as __device__ int*
typedef __attribute__((address_space(3))) int l3_int;   // LDS

static __device__ __forceinline__ void wait_async0()
{
#if __has_builtin(__builtin_amdgcn_s_wait_asynccnt)
    __builtin_amdgcn_s_wait_asynccnt(0);
#else
    asm volatile("s_wait_asynccnt 0x0" ::: "memory");   // SOPP opcode 74
#endif
}
#endif

// ---------------------------------------------------------------------------
// Kernel 1: logits = hidden @ W^T + b, then log_softmax, stored as (T,B,C).
// 4 waves / block, each wave owns a 16-row tile; 3 WMMA accumulators cover
// N = 0..47 (cols 40..47 are zero-masked).
// ---------------------------------------------------------------------------
__global__ __launch_bounds__(128) void gemm_logsoftmax_kernel(
    const float* __restrict__ hidden,   // (B,T,H) row-major -> rows of GEMM
    const float* __restrict__ W,        // (C,H)
    const float* __restrict__ bias,     // (C)
    float* __restrict__ lp)             // (T,B,C)
{
    const int lane  = threadIdx.x & 31;
    const int wave  = threadIdx.x >> 5;
    const int r0    = (blockIdx.x * 4 + wave) * 16;   // first GEMM row of tile
    const int mlo   = lane & 15;
    const int khalf = lane >> 4;                      // 0: K{0,1}, 1: K{2,3}

    // A fragment source: row (r0+mlo), contiguous K pair starting at khalf*2.
    const float* Arow = hidden + (size_t)(r0 + mlo) * H_DIM + khalf * 2;

    // B fragment sources: B[k][n] = W[n][k]; lane -> column n, vgpr -> k pair.
    const int  n0   = mlo;
    const int  n1   = 16 + mlo;
    const int  n2   = 32 + mlo;
    const bool v2ok = (n2 < C_DIM);
    const float* B0 = W + (size_t)n0 * H_DIM + khalf * 2;
    const float* B1 = W + (size_t)n1 * H_DIM + khalf * 2;
    const float* B2 = W + (size_t)(v2ok ? n2 : (C_DIM - 1)) * H_DIM + khalf * 2;

    v8f c0 = {}; v8f c1 = {}; v8f c2 = {};

    #pragma unroll 4
    for (int kk = 0; kk < H_DIM / 4; ++kk) {
        if ((kk & 31) == 0)   // pull the next 512B of this row toward L0
            __builtin_prefetch(Arow + kk * 4 + 128, 0, 1);

        v2f a  = *reinterpret_cast<const v2f*>(Arow + kk * 4);
        v2f b0 = *reinterpret_cast<const v2f*>(B0 + kk * 4);
        v2f b1 = *reinterpret_cast<const v2f*>(B1 + kk * 4);
        v2f b2 = *reinterpret_cast<const v2f*>(B2 + kk * 4);
        if (!v2ok) { b2.x = 0.0f; b2.y = 0.0f; }   // zero padded columns

        c0 = __builtin_amdgcn_wmma_f32_16x16x4_f32(false, a, false, b0,
                                                   (short)0, c0, false, false);
        c1 = __builtin_amdgcn_wmma_f32_16x16x4_f32(false, a, false, b1,
                                                   (short)0, c1, false, false);
        c2 = __builtin_amdgcn_wmma_f32_16x16x4_f32(false, a, false, b2,
                                                   (short)0, c2, false, false);
    }

    const float bz0 = bias[n0];
    const float bz1 = bias[n1];
    const float bz2 = v2ok ? bias[n2] : 0.0f;

    // D layout: VGPR j holds row j (lanes 0-15) / row j+8 (lanes 16-31),
    // lane%16 = column. 16-lane xor reductions stay inside each half-wave,
    // so each row reduces independently.
    #pragma unroll
    for (int j = 0; j < 8; ++j) {
        float x0 = c0[j] + bz0;
        float x1 = c1[j] + bz1;
        float x2 = v2ok ? (c2[j] + bz2) : NEGF;

        float mx = fmaxf(x0, fmaxf(x1, x2));
        #pragma unroll
        for (int off = 1; off < 16; off <<= 1)
            mx = fmaxf(mx, __shfl_xor(mx, off, 32));

        float sm = __expf(x0 - mx) + __expf(x1 - mx)
                 + (v2ok ? __expf(x2 - mx) : 0.0f);
        #pragma unroll
        for (int off = 1; off < 16; off <<= 1)
            sm += __shfl_xor(sm, off, 32);

        const float lz = mx + __logf(sm);

        const int r    = r0 + j + 8 * khalf;       // GEMM row = b*T + t
        const int bidx = r / T_LEN;
        const int t    = r - bidx * T_LEN;
        float* orow = lp + ((size_t)t * NUM_B + bidx) * C_DIM;
        orow[n0] = x0 - lz;
        orow[n1] = x1 - lz;
        if (v2ok) orow[n2] = x2 - lz;
    }
}

// ---------------------------------------------------------------------------
// Kernel 2: CTC alpha recursion. One block per batch element, thread s owns
// extended state s (S=401). alpha ping-pongs in LDS; the 40-wide log-prob
// row for step t+1 is fetched while step t computes, so the global load
// latency never sits on the 2000-step critical path.
// ---------------------------------------------------------------------------
__global__ __launch_bounds__(448) void ctc_alpha_kernel(
    const float* __restrict__ lp,       // (T,B,C)
    const int*   __restrict__ targets,  // (B,L)
    const int*   __restrict__ in_len,   // (B)
    const int*   __restrict__ tg_len,   // (B)
    float*       __restrict__ loss_b)   // (B)
{
    __shared__ float alpha[2][S_EXT];
    __shared__ float row[2][C_DIM];

    const int b   = blockIdx.x;
    const int s   = threadIdx.x;
    const int inl = in_len[b];
    const int tgl = tg_len[b];

    // Per-state label + "skip" (s-2 transition allowed) hoisted to registers.
    int  label = BLANKC;
    bool skip  = false;
    if (s < S_EXT && (s & 1)) {
        const int li = s >> 1;
        label = targets[b * L_TGT + li];
        if (s >= 3) skip = (label != targets[b * L_TGT + li - 1]);
    }

    if (s < S_EXT) {
        float a0 = NEGF;
        if (s == 0)      a0 = lp[(size_t)b * C_DIM + BLANKC];
        else if (s == 1) a0 = lp[(size_t)b * C_DIM + label];
        alpha[0][s] = a0;
    }

#if HAVE_ASYNC_LDS
    // Pre-issue the emission row for t = 1 straight into LDS (ASYNCcnt).
    if (s < C_DIM)
        __builtin_amdgcn_global_load_async_to_lds_b32(
            (g1_int*)(void*)(lp + ((size_t)1 * NUM_B + b) * C_DIM + s),
            (l3_int*)(void*)&row[1][s], 0, 0);
#else
    // Register-pipelined fallback: rnext always holds the row for step t.
    float rnext = (s < C_DIM)
                ? lp[((size_t)1 * NUM_B + b) * C_DIM + s] : 0.0f;
#endif

    int cur = 0;
    for (int t = 1; t < T_LEN; ++t) {
        const int rb = t & 1;

#if HAVE_ASYNC_LDS
        wait_async0();       // row[rb] has landed in LDS
        __syncthreads();     // alpha[cur] final from t-1; row[rb] visible
        if (t + 1 < T_LEN && s < C_DIM)   // readers of row[rb^1] are done
            __builtin_amdgcn_global_load_async_to_lds_b32(
                (g1_int*)(void*)(lp + ((size_t)(t + 1) * NUM_B + b) * C_DIM + s),
                (l3_int*)(void*)&row[rb ^ 1][s], 0, 0);
#else
        if (s < C_DIM)
            row[rb][s] = rnext;           // stage row for step t
        if (t + 1 < T_LEN && s < C_DIM)   // issue fetch for t+1 early
            rnext = lp[((size_t)(t + 1) * NUM_B + b) * C_DIM + s];
        __syncthreads();     // alpha[cur] final from t-1; row[rb] ready
#endif

        if (s < S_EXT) {
            const float a  = alpha[cur][s];
            const float a1 = (s >= 1) ? alpha[cur][s - 1] : NEGF;
            const float a2 = skip     ? alpha[cur][s - 2] : NEGF;
            const float m  = fmaxf(a, fmaxf(a1, a2));
            const float lse = m + __logf(__expf(a - m) + __expf(a1 - m)
                                         + __expf(a2 - m));
            const float nv = lse + row[rb][label];
            alpha[cur ^ 1][s] = (t < inl) ? nv : a;   // freeze past length
        }
        cur ^= 1;
    }
    __syncthreads();

    if (s == 0) {
        const int   e  = 2 * tgl;
        const float ab = alpha[cur][e];
        const float al = alpha[cur][e - 1];
        const float m  = fmaxf(ab, al);
        float loss = -(m + __logf(__expf(ab - m) + __expf(al - m)));
        if (loss > 1e29f) loss = 0.0f;                // zero_infinity
        loss_b[b] = loss / (float)tgl;
    }
}

// ---------------------------------------------------------------------------
// Kernel 3: mean of per-batch losses (single wave32).
// ---------------------------------------------------------------------------
__global__ __launch_bounds__(32) void loss_reduce_kernel(
    const float* __restrict__ loss_b, float* __restrict__ out)
{
    float v = (threadIdx.x < NUM_B) ? loss_b[threadIdx.x] : 0.0f;
    #pragma unroll
    for (int off = 16; off >= 1; off >>= 1)
        v += __shfl_xor(v, off, 32);
    if (threadIdx.x == 0) out[0] = v / (float)NUM_B;
}

// ---------------------------------------------------------------------------
extern "C" void kernel_launch(void* const* d_in, const int* in_sizes, int n_in,
                              void* d_out, int out_size, void* d_ws, size_t ws_size,
                              hipStream_t stream)
{
    const float* hidden  = (const float*)d_in[0];   // (32,2000,512) f32
    const float* W       = (const float*)d_in[1];   // (40,512) f32
    const float* bias    = (const float*)d_in[2];   // (40) f32
    const int*   targets = (const int*)d_in[3];     // (32,200) i32
    const int*   in_len  = (const int*)d_in[4];     // (32) i32
    const int*   tg_len  = (const int*)d_in[5];     // (32) i32

    float* lp     = (float*)d_out;                  // first T*B*C floats
    float* loss   = lp + (size_t)T_LEN * NUM_B * C_DIM;
    float* loss_b = (float*)d_ws;                   // 32 floats of scratch

    // 64000 rows / (4 waves * 16 rows) = 1000 blocks.
    gemm_logsoftmax_kernel<<<1000, 128, 0, stream>>>(hidden, W, bias, lp);
    ctc_alpha_kernel<<<NUM_B, 448, 0, stream>>>(lp, targets, in_len, tg_len, loss_b);
    loss_reduce_kernel<<<1, 32, 0, stream>>>(loss_b, loss);
}